// TAGNodeReg_6399501271545
// MI455X (gfx1250) — compile-verified
//
#include <hip/hip_runtime.h>
#include <math.h>

typedef __attribute__((ext_vector_type(2))) float v2f;
typedef __attribute__((ext_vector_type(8))) float v8f;

#define THREADS 256

static inline unsigned nblk(unsigned n) { return (n + THREADS - 1) / THREADS; }

// ---------------------------------------------------------------------------
// Utility: zero a float buffer (needed before every atomic scatter pass)
// ---------------------------------------------------------------------------
__global__ void zero_f32_kernel(float* __restrict__ p, unsigned n) {
  unsigned i = blockIdx.x * blockDim.x + threadIdx.x;
  if (i < n) p[i] = 0.0f;
}

// ---------------------------------------------------------------------------
// Degree: deg[dst] += w  (one thread per edge)
// ---------------------------------------------------------------------------
__global__ void deg_kernel(const int* __restrict__ dst,
                           const float* __restrict__ w,
                           float* __restrict__ deg, unsigned E) {
  unsigned e = blockIdx.x * blockDim.x + threadIdx.x;
  if (e < E) atomicAdd(&deg[dst[e]], w[e]);
}

// ---------------------------------------------------------------------------
// GCN norm: norm[e] = dinv[src] * w * dinv[dst], dinv = deg>0 ? rsqrt(deg) : 0
// ---------------------------------------------------------------------------
__global__ void norm_kernel(const int* __restrict__ src,
                            const int* __restrict__ dst,
                            const float* __restrict__ w,
                            const float* __restrict__ deg,
                            float* __restrict__ norm, unsigned E) {
  unsigned e = blockIdx.x * blockDim.x + threadIdx.x;
  if (e >= E) return;
  float ds = deg[src[e]];
  float dd = deg[dst[e]];
  float a = ds > 0.0f ? rsqrtf(ds) : 0.0f;
  float b = dd > 0.0f ? rsqrtf(dd) : 0.0f;
  norm[e] = a * w[e] * b;
}

// ---------------------------------------------------------------------------
// conv1 hop-0: acc[i][j] = x[i][0]*W[0][j] + x[i][1]*W[1][j] + bias[j]
// One thread per (node, out-channel); 16 channels.
// ---------------------------------------------------------------------------
__global__ void conv1_init_kernel(const float* __restrict__ x,
                                  const float* __restrict__ W,   // [2][16]
                                  const float* __restrict__ bias,
                                  float* __restrict__ acc, unsigned N) {
  unsigned t = blockIdx.x * blockDim.x + threadIdx.x;
  unsigned i = t >> 4, j = t & 15u;
  if (i >= N) return;
  float x0 = x[(size_t)i * 2 + 0];
  float x1 = x[(size_t)i * 2 + 1];
  acc[(size_t)i * 16 + j] = fmaf(x0, W[j], fmaf(x1, W[16 + j], bias[j]));
}

// ---------------------------------------------------------------------------
// 2-channel propagation: hn[dst] += norm * h[src]   (one thread per edge)
// ---------------------------------------------------------------------------
__global__ void prop2_kernel(const float* __restrict__ h,   // [N,2]
                             float* __restrict__ hn,        // [N,2], pre-zeroed
                             const int* __restrict__ src,
                             const int* __restrict__ dst,
                             const float* __restrict__ norm, unsigned E) {
  unsigned e = blockIdx.x * blockDim.x + threadIdx.x;
  if (e >= E) return;
  int s = src[e], d = dst[e];
  float nv = norm[e];
  float h0 = h[(size_t)s * 2 + 0];
  float h1 = h[(size_t)s * 2 + 1];
  atomicAdd(&hn[(size_t)d * 2 + 0], nv * h0);
  atomicAdd(&hn[(size_t)d * 2 + 1], nv * h1);
}

// ---------------------------------------------------------------------------
// conv1 accumulate: acc[i][j] += h[i][0]*W[0][j] + h[i][1]*W[1][j]
// ---------------------------------------------------------------------------
__global__ void accum2_kernel(const float* __restrict__ h,   // [N,2]
                              const float* __restrict__ W,   // [2][16]
                              float* __restrict__ acc, unsigned N) {
  unsigned t = blockIdx.x * blockDim.x + threadIdx.x;
  unsigned i = t >> 4, j = t & 15u;
  if (i >= N) return;
  float h0 = h[(size_t)i * 2 + 0];
  float h1 = h[(size_t)i * 2 + 1];
  acc[(size_t)i * 16 + j] += fmaf(h0, W[j], h1 * W[16 + j]);
}

// ---------------------------------------------------------------------------
// Elementwise relu: dstbuf = max(srcbuf, 0)
// ---------------------------------------------------------------------------
__global__ void relu_kernel(const float* __restrict__ s, float* __restrict__ d,
                            unsigned n) {
  unsigned i = blockIdx.x * blockDim.x + threadIdx.x;
  if (i < n) d[i] = fmaxf(s[i], 0.0f);
}

// ---------------------------------------------------------------------------
// 16-channel propagation: 16 lanes per edge; gather/scatter are contiguous
// 64B per edge. hn must be pre-zeroed.
// ---------------------------------------------------------------------------
__global__ void prop16_kernel(const float* __restrict__ h,   // [N,16]
                              float* __restrict__ hn,        // [N,16]
                              const int* __restrict__ src,
                              const int* __restrict__ dst,
                              const float* __restrict__ norm, unsigned E) {
  unsigned t = blockIdx.x * blockDim.x + threadIdx.x;
  unsigned e = t >> 4, c = t & 15u;
  if (e >= E) return;
  int s = src[e], d = dst[e];
  float v = norm[e] * h[(size_t)s * 16 + c];
  atomicAdd(&hn[(size_t)d * 16 + c], v);
}

// ---------------------------------------------------------------------------
// WMMA [16-node tile, 16] @ W[16,16] (+bias or +acc) using
// V_WMMA_F32_16X16X4_F32, K split into 4 chunks of 4.
//
// Layouts per CDNA5 ISA 7.12.2 (wave32):
//   A 16x4 : lanes 0-15 -> M=lane, VGPR{0,1}=K{0,1}; lanes 16-31 -> K{2,3}
//   B 4x16 : mirrors A's K split: VGPR r, laneHalf h -> K = r + 2h, N = lane&15
//   C/D    : VGPR r -> M = r + 8*(lane>>4), N = lane&15
// Tile guard is wave-uniform so EXEC is all-ones for every WMMA.
// ---------------------------------------------------------------------------
template <bool INIT>
__global__ void gemm16_wmma_kernel(const float* __restrict__ h,    // [N,16]
                                   const float* __restrict__ W,    // [16,16]
                                   const float* __restrict__ bias, // [16] (INIT)
                                   float* __restrict__ acc,        // [N,16]
                                   unsigned N) {
  unsigned wave = (blockIdx.x * blockDim.x + threadIdx.x) >> 5;
  unsigned lane = threadIdx.x & 31u;
  unsigned nodeBase = wave << 4;               // 16 nodes per wave
  if (nodeBase >= N) return;                   // wave-uniform exit
  unsigned half = lane >> 4;                   // 0 or 1
  unsigned n = lane & 15u;                     // N index / A row (M)

  v8f c;
  if (INIT) {
    float bv = bias[n];
#pragma unroll
    for (int r = 0; r < 8; ++r) c[r] = bv;
  } else {
#pragma unroll
    for (int r = 0; r < 8; ++r)
      c[r] = acc[(size_t)(nodeBase + (unsigned)r + 8u * half) * 16 + n];
  }

#pragma unroll
  for (int j = 0; j < 4; ++j) {
    unsigned kb = 4u * (unsigned)j + 2u * half;          // this lane's K base
    v2f a = *(const v2f*)(h + (size_t)(nodeBase + n) * 16 + kb);
    v2f b;
    b.x = W[kb * 16 + n];
    b.y = W[(kb + 1) * 16 + n];
    c = __builtin_amdgcn_wmma_f32_16x16x4_f32(
        /*neg_a=*/false, a, /*neg_b=*/false, b,
        /*c_mod=*/(short)0, c, /*reuse_a=*/false, /*reuse_b=*/false);
  }

#pragma unroll
  for (int r = 0; r < 8; ++r)
    acc[(size_t)(nodeBase + (unsigned)r + 8u * half) * 16 + n] = c[r];
}

// ---------------------------------------------------------------------------
// Final: y[i] = sigmoid( relu(acc[i]) . Wend )
// ---------------------------------------------------------------------------
__global__ void final_kernel(const float* __restrict__ acc,
                             const float* __restrict__ Wend,
                             float* __restrict__ y, unsigned N) {
  unsigned i = blockIdx.x * blockDim.x + threadIdx.x;
  if (i >= N) return;
  float s = 0.0f;
#pragma unroll
  for (int j = 0; j < 16; ++j)
    s = fmaf(fmaxf(acc[(size_t)i * 16 + j], 0.0f), Wend[j], s);
  y[i] = 1.0f / (1.0f + expf(-s));
}

// ---------------------------------------------------------------------------
// Host driver (graph-capture safe: kernels only, scratch from d_ws)
// ---------------------------------------------------------------------------
extern "C" void kernel_launch(void* const* d_in, const int* in_sizes, int n_in,
                              void* d_out, int out_size, void* d_ws, size_t ws_size,
                              hipStream_t stream) {
  const float* x    = (const float*)d_in[0];   // [N,2]
  const int*   eidx = (const int*)d_in[1];     // [2,E]
  const float* ew   = (const float*)d_in[2];   // [E]
  const float* W1   = (const float*)d_in[3];   // [5,2,16]
  const float* b1   = (const float*)d_in[4];   // [16]
  const float* W2   = (const float*)d_in[5];   // [5,16,16]
  const float* b2   = (const float*)d_in[6];   // [16]
  const float* Wend = (const float*)d_in[7];   // [16,1]
  float* out = (float*)d_out;

  const unsigned N = (unsigned)(in_sizes[0] / 2);
  const unsigned E = (unsigned)in_sizes[2];
  const int* src = eidx;
  const int* dst = eidx + E;

  // Workspace layout (floats): deg[N] | norm[E] | acc[16N] | hA[16N] | hB[16N]
  float* deg  = (float*)d_ws;
  float* norm = deg + N;
  float* acc  = norm + E;
  float* hA   = acc + (size_t)N * 16;
  float* hB   = hA + (size_t)N * 16;

  // ---- gcn_norm ----
  zero_f32_kernel<<<nblk(N), THREADS, 0, stream>>>(deg, N);
  deg_kernel<<<nblk(E), THREADS, 0, stream>>>(dst, ew, deg, E);
  norm_kernel<<<nblk(E), THREADS, 0, stream>>>(src, dst, ew, deg, norm, E);

  // ---- conv1: 2 -> 16, K=4 hops ----
  conv1_init_kernel<<<nblk(N * 16), THREADS, 0, stream>>>(x, W1, b1, acc, N);
  const float* hcur2 = x;          // 2-channel hop state
  float* hping[2] = {hA, hB};      // reuse 16-ch buffers (only front 2N used)
  for (int k = 1; k <= 4; ++k) {
    float* hn = hping[k & 1];
    zero_f32_kernel<<<nblk(N * 2), THREADS, 0, stream>>>(hn, N * 2);
    prop2_kernel<<<nblk(E), THREADS, 0, stream>>>(hcur2, hn, src, dst, norm, E);
    accum2_kernel<<<nblk(N * 16), THREADS, 0, stream>>>(hn, W1 + (size_t)k * 32,
                                                        acc, N);
    hcur2 = hn;
  }

  // ---- relu -> conv2 input ----
  relu_kernel<<<nblk(N * 16), THREADS, 0, stream>>>(acc, hA, N * 16);

  // ---- conv2: 16 -> 16, K=4 hops (WMMA transforms) ----
  const unsigned waves = (N + 15) / 16;
  const unsigned wthreads = waves * 32;
  gemm16_wmma_kernel<true><<<nblk(wthreads), THREADS, 0, stream>>>(
      hA, W2, b2, acc, N);
  float* hc = hA;
  float* hn = hB;
  for (int k = 1; k <= 4; ++k) {
    zero_f32_kernel<<<nblk(N * 16), THREADS, 0, stream>>>(hn, N * 16);
    prop16_kernel<<<nblk(E * 16), THREADS, 0, stream>>>(hc, hn, src, dst, norm, E);
    gemm16_wmma_kernel<false><<<nblk(wthreads), THREADS, 0, stream>>>(
        hn, W2 + (size_t)k * 256, nullptr, acc, N);
    float* t = hc; hc = hn; hn = t;
  }

  // ---- relu + endLinear + sigmoid ----
  final_kernel<<<nblk(N), THREADS, 0, stream>>>(acc, Wend, out, N);
}